// SelfAttention_71674414235763
// MI455X (gfx1250) — compile-verified
//
#include <hip/hip_runtime.h>
#include <hip/hip_bf16.h>
#include <math.h>

#define Lq   2048
#define Hh   16
#define Dd   64
#define Mm   24
#define WIN  128
#define NCH  16
#define CHS  128
#define SCALEF 0.125f
#define DN 0.35355339059327373f      // sqrt(0.125)
#define HALF_LOG_M 1.5890269151739727f  // 0.5*ln(24)
#define EPSF 1.0e-4f

typedef __attribute__((ext_vector_type(2))) float    v2f;
typedef __attribute__((ext_vector_type(8))) float    v8f;
typedef __attribute__((ext_vector_type(4))) unsigned v4u;
typedef __attribute__((ext_vector_type(4))) int      v4i;
typedef __attribute__((ext_vector_type(8))) int      v8i;

__device__ __forceinline__ v8f wmma_f32(v2f a, v2f b, v8f c) {
  // V_WMMA_F32_16X16X4_F32 : D = A(16x4) * B(4x16) + C(16x16), all fp32
  return __builtin_amdgcn_wmma_f32_16x16x4_f32(false, a, false, b, (short)0, c, false, false);
}

__device__ __forceinline__ float hmax16(float v) {
  v = fmaxf(v, __shfl_xor(v, 1));
  v = fmaxf(v, __shfl_xor(v, 2));
  v = fmaxf(v, __shfl_xor(v, 4));
  v = fmaxf(v, __shfl_xor(v, 8));
  return v;
}
__device__ __forceinline__ float hsum16(float v) {
  v += __shfl_xor(v, 1);
  v += __shfl_xor(v, 2);
  v += __shfl_xor(v, 4);
  v += __shfl_xor(v, 8);
  return v;
}
// order-preserving float<->uint encoding for atomicMax
__device__ __forceinline__ unsigned encf(float f) {
  unsigned u = __float_as_uint(f);
  return (u & 0x80000000u) ? ~u : (u | 0x80000000u);
}
__device__ __forceinline__ float decf(unsigned u) {
  unsigned v = (u & 0x80000000u) ? (u & 0x7FFFFFFFu) : ~u;
  return __uint_as_float(v);
}

// ---------------------------------------------------------------- TDM staging
#if __has_builtin(__builtin_amdgcn_tensor_load_to_lds)
#define HAVE_TDM 1
// One TENSOR_LOAD_TO_LDS: 2-D tile (tile_d0 x tile_d1 elements of 4B),
// row stride stride0 (elements), packed contiguously into LDS at lds_dst.
// D# group0/group1 fields per CDNA5 ISA §8.3 / §8.4.
__device__ __forceinline__ void tdm_load_2d(void* lds_dst, const void* gsrc,
                                            unsigned tile_d0, unsigned tile_d1,
                                            unsigned long long stride0,
                                            unsigned tensor_d0, unsigned tensor_d1) {
  unsigned long long ga = (unsigned long long)(uintptr_t)gsrc;   // 57-bit byte addr
  unsigned lds_byte = (unsigned)(uintptr_t)lds_dst;              // LDS = addr[31:0]
  v4u g0;
  g0[0] = 1u;                                                    // count=1, user mode
  g0[1] = lds_byte;                                              // [63:32] lds_addr
  g0[2] = (unsigned)ga;                                          // [95:64] global_addr lo
  g0[3] = (unsigned)((ga >> 32) & 0x01FFFFFFu) | 0x80000000u;    // addr hi | type=2
  v8i g1;
  g1[0] = 0x20000;                                               // data_size=4B, mask=0
  g1[1] = (int)((tensor_d0 & 0xFFFFu) << 16);                    // tensor_dim0[15:0]
  g1[2] = (int)(((tensor_d0 >> 16) & 0xFFFFu) | ((tensor_d1 & 0xFFFFu) << 16));
  g1[3] = (int)(((tensor_d1 >> 16) & 0xFFFFu) | ((tile_d0 & 0xFFFFu) << 16));
  g1[4] = (int)(tile_d1 & 0xFFFFu);                              // tile_dim1, tile_dim2=0
  g1[5] = (int)(unsigned)(stride0 & 0xFFFFFFFFull);              // dim0_stride lo
  g1[6] = (int)(unsigned)((stride0 >> 32) & 0xFFFFull);          // dim0_stride hi
  g1[7] = 0;
  v4i gz = {0, 0, 0, 0};
#if __clang_major__ >= 23
  v8i gz8 = {0, 0, 0, 0, 0, 0, 0, 0};
  __builtin_amdgcn_tensor_load_to_lds(g0, g1, gz, gz, gz8, 0);
#else
  __builtin_amdgcn_tensor_load_to_lds(g0, g1, gz, gz, 0);
#endif
}
#endif

// ---------------------------------------------------------------- init
__global__ void init_kernel(unsigned* stab) {
  if (threadIdx.x < Hh) stab[threadIdx.x] = 0u;  // encodes very negative float
}

// ---------------------------------------------------------------- q features
__global__ __launch_bounds__(256) void feat_q_kernel(
    const float* __restrict__ qkv, const float* __restrict__ proj,
    float* __restrict__ q_prime, float* __restrict__ q_log) {
  __shared__ float pj[Mm * Dd];
  for (int i = threadIdx.x; i < Mm * Dd; i += 256) pj[i] = proj[i];
  __syncthreads();
  int idx = blockIdx.x * 256 + threadIdx.x;       // idx = h*L + l
  int h = idx / Lq, l = idx % Lq;
  const float* qr = qkv + ((size_t)(l * 3 + 0) * Hh + h) * Dd;
  float xs[Dd]; float xn = 0.f;
  for (int d = 0; d < Dd; ++d) { float v = qr[d] * DN; xs[d] = v; xn += v * v; }
  xn *= 0.5f;
  float dash[Mm]; float mx = -1e30f;
  for (int m = 0; m < Mm; ++m) {
    float a = 0.f;
    for (int d = 0; d < Dd; ++d) a += xs[d] * pj[m * Dd + d];
    dash[m] = a; mx = fmaxf(mx, a);
  }
  float* qp = q_prime + (size_t)idx * Mm;
  for (int m = 0; m < Mm; ++m) qp[m] = __expf(dash[m] - mx) + EPSF;
  q_log[idx] = mx - xn - HALF_LOG_M;
}

// ---------------------------------------------------------------- k dash + head max
__global__ __launch_bounds__(256) void feat_kdash_kernel(
    const float* __restrict__ qkv, const float* __restrict__ proj,
    float* __restrict__ k_dash, float* __restrict__ xnormk, unsigned* __restrict__ stab) {
  __shared__ float pj[Mm * Dd];
  for (int i = threadIdx.x; i < Mm * Dd; i += 256) pj[i] = proj[i];
  __syncthreads();
  int idx = blockIdx.x * 256 + threadIdx.x;
  int h = idx / Lq, l = idx % Lq;
  const float* kr = qkv + ((size_t)(l * 3 + 1) * Hh + h) * Dd;
  float xs[Dd]; float xn = 0.f;
  for (int d = 0; d < Dd; ++d) { float v = kr[d] * DN; xs[d] = v; xn += v * v; }
  xn *= 0.5f;
  float mx = -1e30f;
  float* kd = k_dash + (size_t)idx * Mm;
  for (int m = 0; m < Mm; ++m) {
    float a = 0.f;
    for (int d = 0; d < Dd; ++d) a += xs[d] * pj[m * Dd + d];
    kd[m] = a; mx = fmaxf(mx, a);
  }
  xnormk[idx] = xn;
  atomicMax(&stab[h], encf(mx));
}

// ---------------------------------------------------------------- k prime
__global__ __launch_bounds__(256) void feat_kprime_kernel(
    const float* __restrict__ k_dash, const float* __restrict__ xnormk,
    const unsigned* __restrict__ stab, float* __restrict__ k_prime) {
  int idx = blockIdx.x * 256 + threadIdx.x;
  int h = idx / Lq;
  float st = decf(stab[h]);
  float xn = xnormk[idx];
  const float* kd = k_dash + (size_t)idx * Mm;
  float* kp = k_prime + (size_t)idx * Mm;
  for (int m = 0; m < Mm; ++m) kp[m] = __expf(kd[m] - xn - st) + EPSF;
}

// ---------------------------------------------------------------- per-chunk scan sums
__global__ __launch_bounds__(256) void chunk_sums_kernel(
    const float* __restrict__ qkv, const float* __restrict__ k_prime,
    float* __restrict__ s_chunk, float* __restrict__ S_chunk) {
  int h = blockIdx.x / NCH;
  int cc = blockIdx.x % NCH;
  __shared__ float kp[CHS * Mm];
  for (int i = threadIdx.x; i < CHS * Mm; i += 256)
    kp[i] = k_prime[((size_t)h * Lq + cc * CHS + i / Mm) * Mm + (i % Mm)];
  __syncthreads();
  for (int p = threadIdx.x; p < Mm * Dd; p += 256) {
    int m = p >> 6, d = p & 63;
    float acc = 0.f;
    for (int ll = 0; ll < CHS; ++ll) {
      int l = cc * CHS + ll;
      acc += kp[ll * Mm + m] * qkv[((size_t)(l * 3 + 2) * Hh + h) * Dd + d];
    }
    S_chunk[((size_t)(h * NCH + cc) * Mm + m) * Dd + d] = acc;
  }
  if (threadIdx.x < Mm) {
    float a = 0.f;
    for (int ll = 0; ll < CHS; ++ll) a += kp[ll * Mm + threadIdx.x];
    s_chunk[(size_t)(h * NCH + cc) * Mm + threadIdx.x] = a;
  }
}

// ---------------------------------------------------------------- exclusive prefix over chunks
__global__ __launch_bounds__(256) void chunk_prefix_kernel(
    const float* __restrict__ s_chunk, const float* __restrict__ S_chunk,
    float* __restrict__ s_pref, float* __restrict__ S_pref) {
  int h = blockIdx.x;
  for (int p = threadIdx.x; p < Mm * Dd; p += 256) {
    float run = 0.f;
    for (int cc = 0; cc < NCH; ++cc) {
      size_t i = (size_t)(h * NCH + cc) * (Mm * Dd) + p;
      S_pref[i] = run; run += S_chunk[i];
    }
  }
  if (threadIdx.x < Mm) {
    float run = 0.f;
    for (int cc = 0; cc < NCH; ++cc) {
      size_t i = (size_t)(h * NCH + cc) * Mm + threadIdx.x;
      s_pref[i] = run; run += s_chunk[i];
    }
  }
}

// ---------------------------------------------------------------- fused banded + performer attention
__global__ __launch_bounds__(256) void attn_kernel(
    const float* __restrict__ qkv, const float* __restrict__ q_prime,
    const float* __restrict__ k_prime, const float* __restrict__ q_log,
    const unsigned* __restrict__ stab, const float* __restrict__ s_pref,
    const float* __restrict__ S_pref, float* __restrict__ out) {
  extern __shared__ float smem[];
  float* k_s  = smem;                 // 256*64
  float* v_s  = k_s + 256 * Dd;       // 256*64
  float* kp_s = v_s + 256 * Dd;       // 256*24
  float* wscr = kp_s + 256 * Mm;      // 8 waves * 16*17

  const int tid = threadIdx.x;
  const int h = blockIdx.x / NCH;
  const int cidx = blockIdx.x % NCH;
  const int c = cidx * CHS;

  // ---- stage key span [c-128, c+127] into LDS ----
  const int j0 = (c >= 128) ? (c - 128) : 0;
  const int pad = j0 - (c - 128);          // 128 for first block, else 0
  const int nrows = 256 - pad;
#if defined(HAVE_TDM)
  if (tid < 32) {   // wave 0 drives the Tensor Data Mover (EXEC-independent DMA)
    tdm_load_2d(k_s + (size_t)pad * Dd, qkv + ((size_t)(j0 * 3 + 1) * Hh + h) * Dd,
                Dd, (unsigned)nrows, (unsigned long long)(3 * Hh * Dd), Dd, (unsigned)nrows);
    tdm_load_2d(v_s + (size_t)pad * Dd, qkv + ((size_t)(j0 * 3 + 2) * Hh + h) * Dd,
                Dd, (unsigned)nrows, (unsigned long long)(3 * Hh * Dd), Dd, (unsigned)nrows);
    tdm_load_2d(kp_s + (size_t)pad * Mm, k_prime + ((size_t)h * Lq + j0) * Mm,
                Mm, (unsigned)nrows, (unsigned long long)Mm, Mm, (unsigned)nrows);
    __builtin_amdgcn_s_wait_tensorcnt(0);
  }
  if (tid < pad) {   // zero-fill rows before sequence start
    for (int d = 0; d < Dd; ++d) { k_s[tid * Dd + d] = 0.f; v_s[tid * Dd + d] = 0.f; }
    for (int m = 0; m < Mm; ++m) kp_s[tid * Mm + m] = 0.f;
  }
#else
  {
    int j = c - 128 + tid;
    if (j >= 0) {
      const float* kr = qkv + ((size_t)(j * 3 + 1) * Hh + h) * Dd;
      const float* vr = qkv + ((size_t)(j * 3 + 2) * Hh + h) * Dd;
      for (int d = 0; d < Dd; ++d) { k_s[tid * Dd + d] = kr[d]; v_s[tid * Dd + d] = vr[d]; }
      const float* kpr = k_prime + ((size_t)h * Lq + j) * Mm;
      for (int m = 0; m < Mm; ++m) kp_s[tid * Mm + m] = kpr[m];
    } else {
      for (int d = 0; d < Dd; ++d) { k_s[tid * Dd + d] = 0.f; v_s[tid * Dd + d] = 0.f; }
      for (int m = 0; m < Mm; ++m) kp_s[tid * Mm + m] = 0.f;
    }
  }
#endif
  __syncthreads();

  const int lane = tid & 31;
  const int wv = tid >> 5;
  const int hi = lane >> 4;           // upper half-wave
  const int ln16 = lane & 15;
  const int l_base = c + wv * 16;     // wave owns 16 query rows
  const int arow = l_base + ln16;     // A-fragment row for this lane

  // preload A-fragments: q (16 K-steps) and q_prime (6 K-steps)
  v2f qa[16];
  {
    const float* qr = qkv + ((size_t)(arow * 3 + 0) * Hh + h) * Dd;
    for (int t = 0; t < 16; ++t) { qa[t][0] = qr[4 * t + 2 * hi]; qa[t][1] = qr[4 * t + 2 * hi + 1]; }
  }
  v2f qpa[6];
  {
    const float* qpr = q_prime + ((size_t)h * Lq + arow) * Mm;
    for (int t = 0; t < 6; ++t) { qpa[t][0] = qpr[4 * t + 2 * hi]; qpa[t][1] = qpr[4 * t + 2 * hi + 1]; }
  }

  const float klog = decf(stab[h]) - HALF_LOG_M;

  v8f vz = {0, 0, 0, 0, 0, 0, 0, 0};
  v8f U[4] = {vz, vz, vz, vz};   // softmax numerator accumulators (N-tiles of D)
  v8f G[4];                      // linear (performer-correction) accumulators

  // G = q_prime @ S_prefix  (global branch prefix part), via WMMA
  for (int nt = 0; nt < 4; ++nt) {
    v8f acc = vz;
    const float* Sp = S_pref + (size_t)(h * NCH + cidx) * Mm * Dd;
    for (int t = 0; t < 6; ++t) {
      v2f b;
      b[0] = Sp[(4 * t + 2 * hi) * Dd + nt * 16 + ln16];
      b[1] = Sp[(4 * t + 2 * hi + 1) * Dd + nt * 16 + ln16];
      acc = wmma_f32(qpa[t], b, acc);
    }
    G[nt] = acc;
  }

  float rowm[8], rowZ[8], rowGqk[8], rowGoc[8];
  for (int n = 0; n < 8; ++n) { rowm[n] = -1e30f; rowZ[n] = 0.f; rowGqk[n] = 0.f; rowGoc[n] = 0.f; }

  float* wsc = wscr + wv * 16 * 17;   // wave-private C->A relayout scratch

#pragma unroll 1
  for (int ct = 0; ct < 16; ++ct) {
    const int kb = ct * 16;
    const int jg = c - 128 + kb + ln16;   // this lane's key column
    // S = q @ k^T tile
    v8f S = vz;
    for (int t = 0; t < 16; ++t) {
      v2f b;
      b[0] = k_s[(kb + ln16) * Dd + 4 * t + 2 * hi];
      b[1] = k_s[(kb + ln16) * Dd + 4 * t + 2 * hi + 1];
      S = wmma_f32(qa[t], b, S);
    }
    // P = q_prime @ k_prime^T tile
    v8f P = vz;
    for (int t = 0; t < 6; ++t) {
      v2f b;
      b[0] = kp_s[(kb + ln16) * Mm + 4 * t + 2 * hi];
      b[1] = kp_s[(kb + ln16) * Mm + 4 * t + 2 * hi + 1];
      P = wmma_f32(qpa[t], b, P);
    }
    float ev[8], wc[8];
    for (int n = 0; n < 8; ++n) {
      const int l = l_base + n + 8 * hi;
      const int dlt = l - jg;
      const bool valid = (jg >= 0) && (dlt >= 0) && (dlt <= WIN);
      float sv = valid ? S[n] * SCALEF : -1e30f;
      float tm = hmax16(sv);
      float mnew = fmaxf(rowm[n], tm);
      float f = __expf(rowm[n] - mnew);
      float e = valid ? __expf(sv - mnew) : 0.f;
      rowZ[n] = rowZ[n] * f + hsum16(e);
      U[0][n] *= f; U[1][n] *= f; U[2][n] *= f; U[3][n] *= f;
      rowm[n] = mnew;
      ev[n] = e;
      float p = P[n];
      float pa = (valid && (jg < c)) ? p : 0.f;            // band, previous chunk
      float pb = ((jg >= c) && (dlt >= 0)) ? p : 0.f;      // tril, current chunk
      rowGqk[n] += hsum16(valid ? p : 0.f);                // -> gqk_sum (x SCALE later)
      rowGoc[n] += hsum16(pb);                             // within-chunk cumsum part
      wc[n] = (1.f - SCALEF) * pb - SCALEF * pa;           // constant-coefficient combo
    }
    // softmax weights -> U via WMMA (relayout through LDS)
    for (int n = 0; n < 8; ++n) wsc[(n + 8 * hi) * 17 + ln16] = ev[n];
    for (int nt = 0; nt < 4; ++nt) {
      v8f acc = U[nt];
      for (int t = 0; t < 4; ++t) {
        v2f a, b;
        a[0] = wsc[ln16 * 17 + 4 * t + 2 * hi];
        a[1] = wsc[ln16 * 17 + 4 * t + 2 * hi + 1];
        b[0] = v_s[(kb + 4 * t + 2 * hi) * Dd + nt * 16 + ln16];
        b[1] = v_s[(kb + 4 * t + 2 * hi + 1) * Dd + nt * 16 + ln16];
        acc = wmma_f32(a, b, acc);
      }
      U[nt] = acc;
    }
    // combined P weights -> G via WMMA
    for (int n = 0; n < 8; ++n) wsc[(n + 8 * hi) * 17 + ln16] = wc[n];
    for (int nt = 0; nt < 4; ++nt) {
      v8f acc = G[nt];
      for (int t = 0; t < 4; ++t) {
        v2f a, b;
        a[0] = wsc[ln16 * 17 + 4 * t + 2 * hi];
        a[1] = wsc[ln16 * 17 + 4 * t + 2 * hi + 1];
        b[0] = v_s[(kb + 4 * t + 2 * hi) * Dd + nt * 16 + ln16];
        b[1] = v_s[(kb + 4 * t + 2 * hi + 1) * Dd + nt * 16 + ln16];
        acc = wmma_f32(a, b, acc);
      }
      G[nt] = acc;
    }
  }

  // finalize per-row normalizers
  const float* sp = s_pref + (size_t)(h * NCH + cidx) * Mm;
  float gpsv[8], invZ[8];
  for (int n = 0; n < 8; ++n) {
    const int l = l_base + n + 8 * hi;
    float lse = rowm[n] + __logf(rowZ[n]);          // local logsumexp
    float part = 0.f;
    const float* qpr = q_prime + ((size_t)h * Lq + l) * Mm;
    for (int m = ln16; m < Mm; m += 16) part += qpr[m] * sp[m];
    float qps = hsum16(part);                       // q_prime . k_cum prefix
    float goc = qps + rowGoc[n];                    // global_out_cumsum
    float gqk = SCALEF * rowGqk[n];                 // gqk_sum
    float gscale = q_log[h * Lq + l] + klog;        // global_scale
    float gln = __logf(fmaxf(goc - gqk, 1e-24f)) + gscale;
    float lnorm = (lse > gln) ? (lse + log1pf(__expf(gln - lse)))
                              : (gln + log1pf(__expf(lse - gln)));
    gpsv[n] = __expf(gscale - lnorm);               // global_prime_scale
    invZ[n] = 1.0f / rowZ[n];                       // softmax self-normalizes
  }
  for (int nt = 0; nt < 4; ++nt)
    for (int n = 0; n < 8; ++n) {
      const int l = l_base + n + 8 * hi;
      const int d = nt * 16 + ln16;
      out[((size_t)l * Hh + h) * Dd + d] = U[nt][n] * invZ[n] + G[nt][n] * gpsv[n];
    }
}

// ---------------------------------------------------------------- host side
extern "C" void kernel_launch(void* const* d_in, const int* in_sizes, int n_in,
                              void* d_out, int out_size, void* d_ws, size_t ws_size,
                              hipStream_t stream) {
  (void)in_sizes; (void)n_in; (void)out_size; (void)ws_size;
  const float* qkv  = (const float*)d_in[0];
  const float* proj = (const float*)d_in[1];
  float* outp = (float*)d_out;

  float* W = (float*)d_ws;
  size_t off = 0;
  float* q_prime = W + off; off += (size_t)Hh * Lq * Mm;
  float* k_prime = W + off; off += (size_t)Hh * Lq * Mm;
  float* k_dash  = W + off; off += (size_t)Hh * Lq * Mm;
  float* q_log   = W + off; off += (size_t)Hh * Lq;
  float* xnormk  = W + off; off += (size_t)Hh * Lq;
  unsigned* stab = (unsigned*)(W + off); off += 16;
  float* s_chunk = W + off; off += (size_t)Hh * NCH * Mm;
  float* s_pref  = W + off; off += (size_t)Hh * NCH * Mm;
  float* S_chunk = W + off; off += (size_t)Hh * NCH * Mm * Dd;
  float* S_pref  = W + off; off += (size_t)Hh * NCH * Mm * Dd;

  init_kernel<<<1, 64, 0, stream>>>(stab);
  feat_q_kernel<<<(Hh * Lq) / 256, 256, 0, stream>>>(qkv, proj, q_prime, q_log);
  feat_kdash_kernel<<<(Hh * Lq) / 256, 256, 0, stream>>>(qkv, proj, k_dash, xnormk, stab);
  feat_kprime_kernel<<<(Hh * Lq) / 256, 256, 0, stream>>>(k_dash, xnormk, stab, k_prime);
  chunk_sums_kernel<<<Hh * NCH, 256, 0, stream>>>(qkv, k_prime, s_chunk, S_chunk);
  chunk_prefix_kernel<<<Hh, 256, 0, stream>>>(s_chunk, S_chunk, s_pref, S_pref);

  const size_t smem_bytes = (size_t)(256 * Dd + 256 * Dd + 256 * Mm + 8 * 16 * 17) * sizeof(float);
  attn_kernel<<<Hh * NCH, 256, smem_bytes, stream>>>(qkv, q_prime, k_prime, q_log,
                                                     stab, s_pref, S_pref, outp);
}